// Distance_17523466567862
// MI455X (gfx1250) — compile-verified
//
#include <hip/hip_runtime.h>
#include <hip/hip_bf16.h>

#define HD 320
#define WD 320
#define BIGF 1.0e5f

typedef float v2f __attribute__((ext_vector_type(2)));

// ---------------------------------------------------------------------------
// Phase 2 (hot kernel, placed first so it leads the disassembly): one block
// per (batch, row). Async-stage both channels' g2 rows into LDS, build the
// per-row table {c0,c1,x,x} with c = g2[x] + x^2, then the exact lower
// envelope via  d2[j] = j^2 + min_x fma(-2j, x, c[x])
// using v_pk_fma_f32 (both channels per instruction) + v_min3 folding.
// ---------------------------------------------------------------------------
__global__ __launch_bounds__(WD) void edt_phase2(
    const float* __restrict__ g2, float* __restrict__ out)
{
    __shared__ __align__(16) float  raw[2 * WD];   // interleaved raw g2 rows
    __shared__ __align__(16) float4 tab[WD];       // {c0, c1, x, x}

    int b = blockIdx.x / HD;
    int i = blockIdx.x - b * HD;
    int j = threadIdx.x;

    const float* r0 = g2 + ((size_t)(2 * b + 0) * HD + i) * WD;
    const float* r1 = g2 + ((size_t)(2 * b + 1) * HD + i) * WD;

#if defined(__gfx1250__)
    {
        // CDNA5 async copy: each lane copies 4 bytes from the global address in vaddr to the LDS address in vdst, tracked by ASYNCcnt.
        unsigned int l0 = (unsigned int)(unsigned long long)(&raw[2 * j]);
        unsigned int l1 = (unsigned int)(unsigned long long)(&raw[2 * j + 1]);
        asm volatile(
            "global_load_async_to_lds_b32 %0, %1, off\n\t"
            "global_load_async_to_lds_b32 %2, %3, off"
            :
            : "v"(l0), "v"(r0 + j), "v"(l1), "v"(r1 + j)
            : "memory");
    }
# if __has_builtin(__builtin_amdgcn_s_wait_asynccnt)
    __builtin_amdgcn_s_wait_asynccnt(0);
# else
    asm volatile("s_wait_asynccnt 0" ::: "memory");
# endif
#else
    raw[2 * j]     = r0[j];
    raw[2 * j + 1] = r1[j];
#endif
    __syncthreads();

    // Build table entry for x = j: c = g2[x] + x^2 (exact where it matters).
    float xf  = (float)j;
    float xsq = xf * xf;
    tab[j] = make_float4(raw[2 * j] + xsq, raw[2 * j + 1] + xsq, xf, xf);
    __syncthreads();

    // Envelope: s(x) = -2j*x + c[x]; d2 = j^2 + min_x s(x).
    float d0 = 3.0e38f, d1 = 3.0e38f;
    v2f a2;
    a2.x = -2.0f * xf;
    a2.y = a2.x;
    const float4* tv = tab;
#pragma unroll 4
    for (int x = 0; x < WD; x += 2) {
        float4 e0 = tv[x];
        float4 e1 = tv[x + 1];
        v2f c0; c0.x = e0.x; c0.y = e0.y;
        v2f x0; x0.x = e0.z; x0.y = e0.w;
        v2f c1; c1.x = e1.x; c1.y = e1.y;
        v2f x1; x1.x = e1.z; x1.y = e1.w;
        v2f t0, t1;
#if defined(__gfx1250__)
        asm("v_pk_fma_f32 %0, %1, %2, %3" : "=v"(t0) : "v"(a2), "v"(x0), "v"(c0));
        asm("v_pk_fma_f32 %0, %1, %2, %3" : "=v"(t1) : "v"(a2), "v"(x1), "v"(c1));
#else
        t0.x = fmaf(a2.x, x0.x, c0.x); t0.y = fmaf(a2.y, x0.y, c0.y);
        t1.x = fmaf(a2.x, x1.x, c1.x); t1.y = fmaf(a2.y, x1.y, c1.y);
#endif
        d0 = fminf(d0, fminf(t0.x, t1.x));   // -> v_min3
        d1 = fminf(d1, fminf(t0.y, t1.y));
    }
    d0 += xsq;   // + j^2 (exact in the non-underflow range)
    d1 += xsq;

    // denom = 2*sigma^2, sigma = {0.02,0.08,0.16}*320, fp32-folded like the
    // reference ((2*s)*s association). Keep IEEE fp32 division to match jnp.
    const float den0 = (2.0f * (0.02f * 320.0f)) * (0.02f * 320.0f);
    const float den1 = (2.0f * (0.08f * 320.0f)) * (0.08f * 320.0f);
    const float den2 = (2.0f * (0.16f * 320.0f)) * (0.16f * 320.0f);

    float o0 = expf(-d0 / den0);
    float o1 = expf(-d0 / den1);
    float o2 = expf(-d0 / den2);
    float o3 = expf(-d1 / den0);
    float o4 = expf(-d1 / den1);
    float o5 = expf(-d1 / den2);

    // out[b,i,j, c*3+s]; base byte offset is a multiple of 8 -> float2 stores.
    size_t base = (((size_t)(b * HD + i)) * WD + j) * 6;
    float2* op = (float2*)(out + base);
    op[0] = make_float2(o0, o1);
    op[1] = make_float2(o2, o3);
    op[2] = make_float2(o4, o5);
}

// ---------------------------------------------------------------------------
// Phase 1: per-column 1D distance along H (down scan + up scan), g^2 -> ws.
// One thread per (plane, column); planes = B*2 (channel folded into batch).
//   last = running max of (h if zero else -1e5)   (downward)
//   nxt  = running min of (h if zero else  2e5)   (upward)
//   g    = min(min(h-last, nxt-h), 1e5);  g2 = g*g
// ---------------------------------------------------------------------------
__global__ __launch_bounds__(256) void edt_phase1(
    const float* __restrict__ in, float* __restrict__ g2, int nplanes)
{
    int t = blockIdx.x * blockDim.x + threadIdx.x;
    if (t >= nplanes * WD) return;
    int n = t / WD;            // plane = b*2 + c
    int w = t - n * WD;
    int b = n >> 1;
    int c = n & 1;

    const float* col = in + (((size_t)b * HD) * WD + w) * 2 + c;  // + h*WD*2
    float*      grow = g2 + ((size_t)n * HD) * WD + w;            // + h*WD

    // down pass: store (h - last) as intermediate
    float last = -BIGF;
    for (int h = 0; h < HD; ++h) {
        float x  = col[(size_t)h * (WD * 2)];
        float hf = (float)h;
        if ((1.0f - x) * 127.5f < 1.0f) last = hf;   // uint8 cast -> 0
        grow[(size_t)h * WD] = hf - last;
    }
    // up pass: combine with nearest seed below, cap, square
    float nxt = 2.0f * BIGF;
    for (int h = HD - 1; h >= 0; --h) {
        float x  = col[(size_t)h * (WD * 2)];
        float hf = (float)h;
        if ((1.0f - x) * 127.5f < 1.0f) nxt = hf;
        float g = fminf(fminf(grow[(size_t)h * WD], nxt - hf), BIGF);
        grow[(size_t)h * WD] = g * g;
    }
}

// ---------------------------------------------------------------------------
extern "C" void kernel_launch(void* const* d_in, const int* in_sizes, int n_in,
                              void* d_out, int out_size, void* d_ws, size_t ws_size,
                              hipStream_t stream)
{
    const float* in  = (const float*)d_in[0];
    float*       out = (float*)d_out;
    float*       g2  = (float*)d_ws;   // [B*2, 320, 320] fp32 = 3.28 MB

    int B       = in_sizes[0] / (HD * WD * 2);   // 4
    int nplanes = B * 2;
    int tot     = nplanes * WD;

    edt_phase1<<<(tot + 255) / 256, 256, 0, stream>>>(in, g2, nplanes);
    edt_phase2<<<B * HD, WD, 0, stream>>>(g2, out);
}